// GraphConv_84378927497742
// MI455X (gfx1250) — compile-verified
//
#include <hip/hip_runtime.h>
#include <hip/hip_bf16.h>

// ---------------------------------------------------------------------------
// GraphConv (GCN) for MI455X / gfx1250, wave32.
//   deg  = undirected degree + 1
//   h    = x @ W^T + b                       (WMMA f32 16x16x4, exact fp32)
//   out  = scatter_add(h[src] * rsqrt(deg[src]*deg[dst]))  over u->v, v->u, i->i
// ---------------------------------------------------------------------------

typedef float v2f __attribute__((ext_vector_type(2)));
typedef float v8f __attribute__((ext_vector_type(8)));

#define D 128           // D_IN == D_OUT == 128

// --------------------------- kernel 1: zero deg -----------------------------
__global__ __launch_bounds__(256) void gc_zero_deg(int* __restrict__ degc, int n) {
    int i = blockIdx.x * 256 + threadIdx.x;
    if (i < n) degc[i] = 0;
}

// --------------------------- kernel 2: degree count -------------------------
__global__ __launch_bounds__(256) void gc_degree(const int* __restrict__ ei,
                                                 int* __restrict__ degc, int E) {
    int e = blockIdx.x * 256 + threadIdx.x;
    if (e >= E) return;
    int u = ei[e];
    int v = ei[E + e];
    __hip_atomic_fetch_add(&degc[u], 1, __ATOMIC_RELAXED, __HIP_MEMORY_SCOPE_AGENT);
    __hip_atomic_fetch_add(&degc[v], 1, __ATOMIC_RELAXED, __HIP_MEMORY_SCOPE_AGENT);
}

// --------------------------- kernel 3: WMMA GEMM ----------------------------
// h[N,128] = x[N,128] @ W[128,128]^T + b.   One wave per 16-row strip; the wave
// computes all 8 column tiles (16x128) with v_wmma_f32_16x16x4_f32.
// A layout (16x4 f32, ISA 7.12.2): lane = M (0..15, repeated), lane half picks
// K+0/K+1 vs K+2/K+3 in the 2 A VGPRs. B (4x16) mirrors this with lane = N.
// C/D: VGPR v holds M=v (lanes 0-15) / M=v+8 (lanes 16-31), N = lane&15.
__global__ __launch_bounds__(256) void gc_gemm(const float* __restrict__ x,
                                               const float* __restrict__ W,
                                               const float* __restrict__ bias,
                                               float* __restrict__ h,
                                               int nrows) {
    __shared__ float Wlds[D * D];          // 64 KB of 320 KB WGP LDS
    const int tid  = threadIdx.x;
    const int lane = tid & 31;
    const int wave = tid >> 5;

    // Stage W into LDS once per block (coalesced), shared by 8 waves.
    #pragma unroll
    for (int i = tid; i < D * D; i += 256) Wlds[i] = W[i];
    __syncthreads();

    const int strip = blockIdx.x * 8 + wave;       // 16-row strip id
    if (strip * 16 >= nrows) return;               // wave-uniform: EXEC stays all-1s

    const int m  = lane & 15;                      // M (A) / N (B) index
    const int kh = (lane >> 4) << 1;               // K sub-offset: 0 or 2
    const float* xrow = x + (size_t)(strip * 16 + m) * D;

    v8f acc[8];
    #pragma unroll
    for (int t = 0; t < 8; ++t) acc[t] = (v8f){0.f, 0.f, 0.f, 0.f, 0.f, 0.f, 0.f, 0.f};

    for (int k0 = 0; k0 < D; k0 += 4) {
        // A tile: x rows, K = k0+kh, k0+kh+1  (8-byte aligned)
        v2f a = *(const v2f*)(xrow + k0 + kh);
        #pragma unroll
        for (int t = 0; t < 8; ++t) {
            // B tile: B[k][n] = W[n][k], n = t*16 + m
            v2f bm = *(const v2f*)(&Wlds[(t * 16 + m) * D + k0 + kh]);
            acc[t] = __builtin_amdgcn_wmma_f32_16x16x4_f32(
                false, a, false, bm, (short)0, acc[t], false, false);
        }
    }

    // Store h tile + bias.
    const int moff = (lane >> 4) << 3;             // +8 rows for upper lane half
    #pragma unroll
    for (int t = 0; t < 8; ++t) {
        const int   col = t * 16 + m;
        const float bc  = bias[col];
        #pragma unroll
        for (int v = 0; v < 8; ++v) {
            int r = strip * 16 + v + moff;
            h[(size_t)r * D + col] = acc[t][v] + bc;
        }
    }
}

// --------------------------- kernel 4: self loops ---------------------------
// out[i,:] = h[i,:] * rsqrt(deg_i*deg_i) = h[i,:] / deg_i.  Plain store -> also
// initializes out (d_out is poisoned by the harness).
__global__ __launch_bounds__(256) void gc_self(const float* __restrict__ h,
                                               const int* __restrict__ degc,
                                               float* __restrict__ out,
                                               long long total) {
    long long i = (long long)blockIdx.x * 256 + threadIdx.x;
    if (i >= total) return;
    int node = (int)(i >> 7);
    float deg = (float)(degc[node] + 1);
    out[i] = h[i] / deg;
}

// --------------------------- kernel 5: edge scatter -------------------------
// One wave per directed message (u->v and v->u per edge); lane owns 4 columns.
// Coalesced 512 B gather of h[src], 4 global_atomic_add_f32 into out[dst].
__global__ __launch_bounds__(256) void gc_edge(const float* __restrict__ h,
                                               const int* __restrict__ ei,
                                               const int* __restrict__ degc,
                                               float* __restrict__ out, int E) {
    const int lane = threadIdx.x & 31;
    const long long msg = (((long long)blockIdx.x * 256) + threadIdx.x) >> 5;
    if (msg >= 2LL * E) return;                    // wave-uniform
    const int e = (int)(msg >> 1);
    const int u = ei[e];
    const int v = ei[E + e];
    const int src = (msg & 1) ? v : u;
    const int dst = (msg & 1) ? u : v;
    const float du = (float)(degc[u] + 1);
    const float dv = (float)(degc[v] + 1);
    const float norm = rsqrtf(du * dv);            // deg >= 1, clip is a no-op

    const float4 hv = *(const float4*)(h + (size_t)src * D + lane * 4);
    float* o = out + (size_t)dst * D + lane * 4;
    __hip_atomic_fetch_add(o + 0, hv.x * norm, __ATOMIC_RELAXED, __HIP_MEMORY_SCOPE_AGENT);
    __hip_atomic_fetch_add(o + 1, hv.y * norm, __ATOMIC_RELAXED, __HIP_MEMORY_SCOPE_AGENT);
    __hip_atomic_fetch_add(o + 2, hv.z * norm, __ATOMIC_RELAXED, __HIP_MEMORY_SCOPE_AGENT);
    __hip_atomic_fetch_add(o + 3, hv.w * norm, __ATOMIC_RELAXED, __HIP_MEMORY_SCOPE_AGENT);
}

// ---------------------------------------------------------------------------
extern "C" void kernel_launch(void* const* d_in, const int* in_sizes, int n_in,
                              void* d_out, int out_size, void* d_ws, size_t ws_size,
                              hipStream_t stream) {
    const float* x    = (const float*)d_in[0];   // [N,128]
    const int*   ei   = (const int*)d_in[1];     // [2,E] (int32 per harness contract)
    const float* W    = (const float*)d_in[2];   // [128,128]
    const float* bias = (const float*)d_in[3];   // [128]
    float*       out  = (float*)d_out;           // [N,128]

    const int N = in_sizes[0] / D;
    const int E = in_sizes[1] / 2;

    // Workspace layout: [ deg counts: N ints | pad to 512 | h: N*128 floats ]
    char*  ws   = (char*)d_ws;
    int*   degc = (int*)ws;
    size_t hoff = ((size_t)N * sizeof(int) + 511) & ~(size_t)511;
    float* h    = (float*)(ws + hoff);
    (void)ws_size; (void)n_in; (void)out_size;

    const long long total = (long long)N * D;

    gc_zero_deg<<<(N + 255) / 256, 256, 0, stream>>>(degc, N);
    gc_degree<<<(E + 255) / 256, 256, 0, stream>>>(ei, degc, E);

    const int nstrips = (N + 15) / 16;                 // 6250 for N=100000
    gc_gemm<<<(nstrips + 7) / 8, 256, 0, stream>>>(x, W, bias, h, N);

    gc_self<<<(unsigned)((total + 255) / 256), 256, 0, stream>>>(h, degc, out, total);

    const long long msgs = 2LL * E;                    // 8 waves (msgs) per block
    gc_edge<<<(unsigned)((msgs + 7) / 8), 256, 0, stream>>>(h, ei, degc, out, E);
}